// NetAndTexture_58050777972780
// MI455X (gfx1250) — compile-verified
//
#include <hip/hip_runtime.h>

typedef __attribute__((ext_vector_type(2))) float v2f;
typedef __attribute__((ext_vector_type(4))) float v4f;
typedef __attribute__((ext_vector_type(8))) float v8f;

#define HH 768
#define WWI 768
#define HWSZ (HH * WWI)          // 589824
#define NPIX (2 * HWSZ)          // 1179648
#define INV_R2 (1.0f / (0.006f * 0.006f))

// One wave (32 lanes) per tile of 16 pixels.
//   lanes 0-15  : pixel q = lane, produce channels 0-7  (point features * mask * weight)
//   lanes 16-31 : pixel q = lane-16, produce channels 8-15 (barycentric face blend * mask)
// LDS layout: sfeat[pixel16][chan16][j12]  (SH dim padded 9 -> 12 so K=192 = 48 WMMA chunks of 4)
// WMMA: D(16px x 16ch) = A(16 x 192 block-diag basis) * B(192 x 16 coeffs)
__global__ __launch_bounds__(32) void NetAndTexture_kernel(
    const float* __restrict__ features,        // (P,72) f32
    const float* __restrict__ frag1_dists,     // (2,H,W,1) f32
    const float* __restrict__ frag2_bary,      // (2,H,W,1,3) f32
    const float* __restrict__ ray_dirs,        // (3,H,W) f32
    const long long* __restrict__ faces,       // (F,3) i64
    const long long* __restrict__ frag1_idx,   // (2,H,W,1) i64
    const long long* __restrict__ frag2_p2f,   // (2,H,W,1) i64
    float* __restrict__ out)                   // [2*H*W*16 feat][2*H*W mask]
{
    __shared__ float sfeat[16 * 192];

    const int lane = threadIdx.x;
    const int q  = lane & 15;   // pixel within tile / A row / D column (channel)
    const int hf = lane >> 4;   // lane half
    const unsigned pix = blockIdx.x * 16u + (unsigned)q;

    // ---- decode pixel -> (layer, h, w) and flipped ray dir ----
    const unsigned layer = (pix >= (unsigned)HWSZ) ? 1u : 0u;
    const unsigned rem = pix - layer * (unsigned)HWSZ;
    const unsigned h  = rem / (unsigned)WWI;
    const unsigned wc = rem - h * (unsigned)WWI;
    const unsigned db = (unsigned)(HH - 1 - (int)h) * (unsigned)WWI + (unsigned)(WWI - 1 - (int)wc);
    const float x = ray_dirs[db];
    const float y = ray_dirs[HWSZ + db];
    const float z = ray_dirs[2 * HWSZ + db];

    // ---- SH degree-2 basis (9 values; j=9..11 are zero pad) ----
    const float xx = x * x, yy = y * y, zz = z * z;
    const float bb0 = 0.28209479177387814f;
    const float bb1 = -0.4886025119029199f * y;
    const float bb2 =  0.4886025119029199f * z;
    const float bb3 = -0.4886025119029199f * x;
    const float bb4 =  1.0925484305920792f * (x * y);
    const float bb5 = -1.0925484305920792f * (y * z);
    const float bb6 =  0.31539156525252005f * (2.0f * zz - xx - yy);
    const float bb7 = -1.0925484305920792f * (x * z);
    const float bb8 =  0.5462742152960396f * (xx - yy);

    // Per-lane A-operand values for the three j-subchunks r=0,1,2 of this pixel's block.
    // A 16x4 layout: vgpr0 holds K = 2*hf, vgpr1 holds K = 2*hf+1 (lane row M = lane&15).
    const float e0r0 = hf ? bb2 : bb0;
    const float e1r0 = hf ? bb3 : bb1;
    const float e0r1 = hf ? bb6 : bb4;
    const float e1r1 = hf ? bb7 : bb5;
    const float e0r2 = hf ? 0.0f : bb8;
    const float e1r2 = 0.0f;

    // ---- masks / indices ----
    const long long i1 = frag1_idx[pix];
    const long long pf = frag2_p2f[pix];
    const bool m1 = (i1 >= 0);
    const bool m2 = (pf >= 0);

    // ---- produce coefficient halves into LDS ----
    float* pb = sfeat + q * 192 + hf * 96;   // this half's 8 channels (12 floats each)

    // zero the j=9..11 pads for our 8 channels
    #pragma unroll
    for (int c = 0; c < 8; ++c) {
        pb[c * 12 + 9]  = 0.0f;
        pb[c * 12 + 10] = 0.0f;
        pb[c * 12 + 11] = 0.0f;
    }

    if (hf == 0) {
        // channels 0-7: features[frag1_idx] * m1 * (1 - dist/R^2)
        const float s1 = m1 ? (1.0f - frag1_dists[pix] * INV_R2) : 0.0f;
        const long long row = m1 ? i1 : 0ll;
        const v4f* src = (const v4f*)(features + row * 72);
        #pragma unroll
        for (int k = 0; k < 18; ++k) {
            const v4f f = src[k];
            #pragma unroll
            for (int i = 0; i < 4; ++i) {
                const int fidx = 4 * k + i;                 // 0..71, compile-time
                pb[(fidx / 9) * 12 + (fidx % 9)] = f[i] * s1;
            }
        }
        // final mask (float 0/1), one lane per pixel
        out[(size_t)(2 * HWSZ) * 16 + pix] = (m1 || m2) ? 1.0f : 0.0f;
    } else {
        // channels 8-15: sum_v bary[v] * features[faces[p2f][v]], masked by m2
        const long long fb = (m2 ? pf : 0ll) * 3;
        const long long t0 = faces[fb + 0];
        const long long t1 = faces[fb + 1];
        const long long t2 = faces[fb + 2];
        float w0 = frag2_bary[(size_t)pix * 3 + 0];
        float w1 = frag2_bary[(size_t)pix * 3 + 1];
        float w2 = frag2_bary[(size_t)pix * 3 + 2];
        if (!m2) { w0 = 0.0f; w1 = 0.0f; w2 = 0.0f; }
        const v4f* r0 = (const v4f*)(features + t0 * 72);
        const v4f* r1 = (const v4f*)(features + t1 * 72);
        const v4f* r2 = (const v4f*)(features + t2 * 72);
        #pragma unroll
        for (int k = 0; k < 18; ++k) {
            const v4f a0 = r0[k];
            const v4f a1 = r1[k];
            const v4f a2 = r2[k];
            #pragma unroll
            for (int i = 0; i < 4; ++i) {
                const int fidx = 4 * k + i;
                pb[(fidx / 9) * 12 + (fidx % 9)] = w0 * a0[i] + w1 * a1[i] + w2 * a2[i];
            }
        }
    }

    __syncthreads();   // single-wave workgroup: barrier is a NOP, waits order LDS

    // ---- 48 chained V_WMMA_F32_16X16X4_F32, block-diagonal A ----
    // B 4x16 layout: vgpr0 = row K=2*hf (col = lane&15), vgpr1 = row K=2*hf+1
    // -> one ds_load_b64 of {j0, j0+1}, j0 = 4r + 2hf, from pixel pt's channel q.
    const float* bsrc = sfeat + q * 12 + hf * 2;
    v8f acc0 = {};
    v8f acc1 = {};
    #pragma unroll
    for (int t = 0; t < 48; ++t) {
        const int pt = t / 3;       // pixel owning this K chunk (compile-time)
        const int r  = t % 3;       // j sub-chunk (compile-time)
        const bool on = (q == pt);  // A row lane is on the diagonal block?
        v2f a;
        a.x = on ? (r == 0 ? e0r0 : (r == 1 ? e0r1 : e0r2)) : 0.0f;
        a.y = on ? (r == 0 ? e1r0 : (r == 1 ? e1r1 : e1r2)) : 0.0f;
        const v2f bv = *(const v2f*)(bsrc + pt * 192 + r * 4);
        if (t & 1)
            acc1 = __builtin_amdgcn_wmma_f32_16x16x4_f32(false, a, false, bv,
                                                         (short)0, acc1, false, false);
        else
            acc0 = __builtin_amdgcn_wmma_f32_16x16x4_f32(false, a, false, bv,
                                                         (short)0, acc0, false, false);
    }
    const v8f res = acc0 + acc1;

    // ---- store D: VGPR r holds (pixel = r + 8*hf, channel = lane&15) ----
    const unsigned opix = blockIdx.x * 16u + 8u * (unsigned)hf;
    #pragma unroll
    for (int r = 0; r < 8; ++r) {
        out[(size_t)(opix + r) * 16 + q] = res[r];
    }
}

extern "C" void kernel_launch(void* const* d_in, const int* in_sizes, int n_in,
                              void* d_out, int out_size, void* d_ws, size_t ws_size,
                              hipStream_t stream) {
    const float*     features = (const float*)d_in[0];
    const float*     dists    = (const float*)d_in[1];
    const float*     bary     = (const float*)d_in[2];
    const float*     rdirs    = (const float*)d_in[3];
    const long long* faces    = (const long long*)d_in[4];
    const long long* f1idx    = (const long long*)d_in[5];
    const long long* p2f      = (const long long*)d_in[6];
    float* out = (float*)d_out;

    const int tiles = NPIX / 16;   // 73728 tiles, one wave each
    hipLaunchKernelGGL(NetAndTexture_kernel, dim3(tiles), dim3(32), 0, stream,
                       features, dists, bary, rdirs, faces, f1idx, p2f, out);
}